// HeatmapProjection_5282809774754
// MI455X (gfx1250) — compile-verified
//
#include <hip/hip_runtime.h>

// ---------------- problem constants (fixed by reference) ----------------
#define SZ        256
#define NPIX      65536              // SZ*SZ
#define NB        32
#define NL        30
#define NBL       960                // NB*NL
#define TOTAL_PIX 62914560ull        // NBL * NPIX

typedef float v2f __attribute__((ext_vector_type(2)));
typedef float v4f __attribute__((ext_vector_type(4)));
typedef float v8f __attribute__((ext_vector_type(8)));

#define LOG2E      1.44269504088896340736f
#define HEAT_NORM  0.006366197723675814f      // 1/(2*pi*25)
#define HEAT_COEF  (-0.02f)                   // -1/(2*VAR), VAR=25
#define SEP_COEF   (-1310.72f)                // -1/(2*var_n), var_n = 25/65536
#define INV_S      (1.0f / 256.0f)

// ======================================================================
// Kernel 1: per-(b,l) softmax moments -> mu_y, mu_x, conc
// 960 blocks x 256 threads; 2 passes over a 256KB slab (pass 2 hits L2)
// ======================================================================
__global__ void __launch_bounds__(256)
k_reduce(const float* __restrict__ y,
         float* __restrict__ muy, float* __restrict__ mux,
         float* __restrict__ conc)
{
    __shared__ float r0[256], r1[256], r2[256], r3[256];
    const int bl = blockIdx.x;
    const int t  = threadIdx.x;
    const v4f* p4 = (const v4f*)(y + (size_t)bl * NPIX);   // 16384 float4s

    // ---- pass A: max ----
    float m = -3.0e38f;
#pragma unroll 4
    for (int i = 0; i < 64; ++i) {
        v4f v = p4[i * 256 + t];
        m = fmaxf(m, fmaxf(fmaxf(v.x, v.y), fmaxf(v.z, v.w)));
    }
    r0[t] = m;
    __syncthreads();
    for (int s = 128; s > 0; s >>= 1) {
        if (t < s) r0[t] = fmaxf(r0[t], r0[t + s]);
        __syncthreads();
    }
    m = r0[0];
    __syncthreads();

    // ---- pass B: moments  Σe, Σe*row, Σe*col, Σe*(row²+col²) ----
    float se = 0.f, sey = 0.f, sex = 0.f, sr2 = 0.f;
#pragma unroll 2
    for (int i = 0; i < 64; ++i) {
        const int q = i * 256 + t;                 // float4 index in slab
        v4f v = p4[q];
        const float row = (float)(q >> 6);         // (4q)>>8
        const float c0  = (float)((q << 2) & 255); // col of first lane-elem
        const float c1 = c0 + 1.f, c2 = c0 + 2.f, c3 = c0 + 3.f;
        const float e0 = __expf(v.x - m);
        const float e1 = __expf(v.y - m);
        const float e2 = __expf(v.z - m);
        const float e3 = __expf(v.w - m);
        const float e  = (e0 + e1) + (e2 + e3);
        se  += e;
        sey += e * row;
        sex += e0 * c0 + e1 * c1 + e2 * c2 + e3 * c3;
        sr2 += e * (row * row)
             + (e0 * c0 * c0 + e1 * c1 * c1 + e2 * c2 * c2 + e3 * c3 * c3);
    }
    r0[t] = se; r1[t] = sey; r2[t] = sex; r3[t] = sr2;
    __syncthreads();
    for (int s = 128; s > 0; s >>= 1) {
        if (t < s) {
            r0[t] += r0[t + s]; r1[t] += r1[t + s];
            r2[t] += r2[t + s]; r3[t] += r3[t + s];
        }
        __syncthreads();
    }
    if (t == 0) {
        const float inv = 1.0f / r0[0];
        const float my = r1[0] * inv;
        const float mx = r2[0] * inv;
        muy[bl] = my;
        mux[bl] = mx;
        // conc = 0.5/S² * (E[y²+x²] − μy² − μx²)
        conc[bl] = 0.5f * (r3[0] * inv - my * my - mx * mx) * (1.0f / 65536.0f);
    }
}

// ======================================================================
// Kernel 2: heatmap synthesis (pure generate; never touches y)
// 61440 blocks x 256 threads; each thread writes one float4 (NT stores)
// ======================================================================
__global__ void __launch_bounds__(256)
k_heatmap(const float* __restrict__ muy, const float* __restrict__ mux,
          v4f* __restrict__ out)
{
    const int blk = blockIdx.x;
    const int bl  = blk >> 6;                       // (b,l) pair
    const int q   = ((blk & 63) << 8) + threadIdx.x; // float4 idx in image
    const float my = muy[bl];                        // block-uniform -> s_load
    const float mx = mux[bl];
    const float row = (float)(q >> 6);
    const float c0  = (float)((q << 2) & 255);
    const float dy  = row - my;
    const float dy2 = dy * dy;
    const float dx0 = c0 - mx;
    const float dx1 = dx0 + 1.f, dx2 = dx0 + 2.f, dx3 = dx0 + 3.f;
    v4f o;
    o.x = HEAT_NORM * __expf((dy2 + dx0 * dx0) * HEAT_COEF);
    o.y = HEAT_NORM * __expf((dy2 + dx1 * dx1) * HEAT_COEF);
    o.z = HEAT_NORM * __expf((dy2 + dx2 * dx2) * HEAT_COEF);
    o.w = HEAT_NORM * __expf((dy2 + dx3 * dx3) * HEAT_COEF);
    __builtin_nontemporal_store(o, out + (size_t)bl * 16384 + q);
}

// ======================================================================
// Kernel 3: prior loss. Gram matrices G = H·Hᵀ via V_WMMA_F32_16X16X4_F32.
// Per batch: H is 30x2 padded to 32x4 (zeros) -> 2x2 tiles of 16x16,
// 128 tiles total, 32 waves x 4 tiles each, single 1024-thread block.
// ======================================================================
__global__ void __launch_bounds__(1024)
k_loss(const float* __restrict__ muy, const float* __restrict__ mux,
       const float* __restrict__ conc, float* __restrict__ out_loss)
{
    __shared__ float hn[NBL];      // |h_i|² per (b,l)
    __shared__ float red[1024];
    const int t = threadIdx.x;

    float cloc = 0.f;
    if (t < NBL) {
        const float a = muy[t] * INV_S;
        const float b = mux[t] * INV_S;
        hn[t] = a * a + b * b;
        cloc  = conc[t];
    }
    __syncthreads();

    const int wave = t >> 5;        // 0..31
    const int lane = t & 31;
    const int li   = lane & 15;
    const bool lowhalf = lane < 16;

    float sep = 0.f;
    for (int k = 0; k < 4; ++k) {                 // wave-uniform loop
        const int tile = wave * 4 + k;            // 0..127
        const int b = tile >> 2;
        const int I = (tile >> 1) & 1;
        const int J = tile & 1;

        // A: 16x4 rows of H (K0,K1 = h_y,h_x in low lanes; K2,K3 = 0)
        // B: 4x16 = Hᵀ (K0 row in VGPR0 low lanes, K1 row in VGPR1 low lanes)
        v2f A = {0.f, 0.f};
        v2f Bv = {0.f, 0.f};
        const int ia = I * 16 + li;
        const int ja = J * 16 + li;
        if (lowhalf && ia < NL) {
            A.x = muy[b * NL + ia] * INV_S;       // K=0
            A.y = mux[b * NL + ia] * INV_S;       // K=1
        }
        if (lowhalf && ja < NL) {
            Bv.x = muy[b * NL + ja] * INV_S;
            Bv.y = mux[b * NL + ja] * INV_S;
        }
        v8f C = {0.f, 0.f, 0.f, 0.f, 0.f, 0.f, 0.f, 0.f};
        v8f D = __builtin_amdgcn_wmma_f32_16x16x4_f32(
            false, A, false, Bv, (short)0, C, false, false);

        const int nj  = J * 16 + li;              // column of this lane
        const float hnj = (nj < NL) ? hn[b * NL + nj] : 0.f;
        const int mbase = I * 16 + (lowhalf ? 0 : 8);
#pragma unroll
        for (int r = 0; r < 8; ++r) {
            const int mi = mbase + r;
            if (mi < NL && nj < NL && mi != nj) {
                float d2 = hn[b * NL + mi] + hnj - 2.f * D[r];
                d2 = fmaxf(d2, 0.f);
                sep += 0.5f * __expf(d2 * SEP_COEF);
            }
        }
    }

    // reduce sep over block
    red[t] = sep;
    __syncthreads();
    for (int s = 512; s > 0; s >>= 1) {
        if (t < s) red[t] += red[t + s];
        __syncthreads();
    }
    const float septot = red[0];
    __syncthreads();
    // reduce conc over block
    red[t] = cloc;
    __syncthreads();
    for (int s = 512; s > 0; s >>= 1) {
        if (t < s) red[t] += red[t + s];
        __syncthreads();
    }
    if (t == 0)
        out_loss[0] = septot * (1.0f / 32.0f) + red[0] * (1.0f / 960.0f);
}

// ======================================================================
extern "C" void kernel_launch(void* const* d_in, const int* in_sizes, int n_in,
                              void* d_out, int out_size, void* d_ws, size_t ws_size,
                              hipStream_t stream)
{
    (void)in_sizes; (void)n_in; (void)out_size; (void)ws_size;
    const float* y  = (const float*)d_in[0];
    float* out      = (float*)d_out;
    float* ws       = (float*)d_ws;
    float* muy  = ws;
    float* mux  = ws + NBL;
    float* conc = ws + 2 * NBL;

    k_reduce<<<NBL, 256, 0, stream>>>(y, muy, mux, conc);
    k_heatmap<<<NBL * 64, 256, 0, stream>>>(muy, mux, (v4f*)out);
    k_loss<<<1, 1024, 0, stream>>>(muy, mux, conc, out + TOTAL_PIX);
}